// Model_54863912239638
// MI455X (gfx1250) — compile-verified
//
#include <hip/hip_runtime.h>
#include <hip/hip_bf16.h>
#include <math.h>

#define NUU 50000
#define NJJ 20000
#define NEDGE 500000
#define NEL 100000
#define DIM 128

typedef __attribute__((ext_vector_type(2))) float v2f;
typedef __attribute__((ext_vector_type(8))) float v8f;

// -------------------- utility kernels --------------------

__global__ __launch_bounds__(256) void k_zero(float* __restrict__ p, int n) {
  int i = blockIdx.x * 256 + threadIdx.x;
  if (i < n) p[i] = 0.0f;
}

__global__ __launch_bounds__(256) void k_count(const int* __restrict__ dst,
                                               float* __restrict__ cnt, int n) {
  int e = blockIdx.x * 256 + threadIdx.x;
  if (e < n) unsafeAtomicAdd(&cnt[dst[e]], 1.0f);
}

// one wave per edge; lane handles 4 consecutive features (32*4 = 128)
__global__ __launch_bounds__(256) void k_scatter(const float* __restrict__ xsrc,
                                                 const int* __restrict__ src,
                                                 const int* __restrict__ dst,
                                                 float* __restrict__ sum, int n) {
  int e = blockIdx.x * 8 + (threadIdx.x >> 5);
  if (e >= n) return;
  int lane = threadIdx.x & 31;
  int s = src[e];
  int d = dst[e];
  const float4 v = *(const float4*)(xsrc + (size_t)s * DIM + lane * 4);
  float* p = sum + (size_t)d * DIM + lane * 4;
  unsafeAtomicAdd(p + 0, v.x);
  unsafeAtomicAdd(p + 1, v.y);
  unsafeAtomicAdd(p + 2, v.z);
  unsafeAtomicAdd(p + 3, v.w);
}

// sum -> mean in place: sum[i] /= max(cnt[i/128], 1)
__global__ __launch_bounds__(256) void k_mean(float* __restrict__ sum,
                                              const float* __restrict__ cnt, int n) {
  int i = blockIdx.x * 256 + threadIdx.x;
  if (i < n) sum[i] = sum[i] / fmaxf(cnt[i >> 7], 1.0f);
}

// -------------------- SAGE GEMM via V_WMMA_F32_16X16X4_F32 --------------------
// out[n,h] = relu?( mean[n,:] @ Wl[h,:] + b[h] + xdst[n,:] @ Wr[h,:] )
// One wave computes a 16x16 output tile; 8 waves/block share a 16-row stripe.
// A fragment (16x4 f32): lanes 0-15 -> M=0..15 K={kk,kk+1}; lanes 16-31 -> K={kk+2,kk+3}
// B fragment (4x16 f32): lanes map N, halves split K, B[k][n] = W[n*128+k]
// C/D (16x16 f32, 8 VGPRs): vgpr r, lane<16 -> (M=r, N=lane); lane>=16 -> (M=r+8, N=lane-16)
// Two independent accumulator chains (Wl product / Wr product) interleaved for ILP;
// bias folded into accA init (every acc VGPR maps to the same output column per lane).
__global__ __launch_bounds__(256) void k_sage_gemm(
    const float* __restrict__ mean, const float* __restrict__ xdst,
    const float* __restrict__ Wl, const float* __restrict__ Wr,
    const float* __restrict__ bias, float* __restrict__ out,
    int nRowTiles, int relu) {
  int wave = threadIdx.x >> 5;
  int lane = threadIdx.x & 31;
  int tile = blockIdx.x * 8 + wave;        // tiles = nRowTiles * 8 (8 col tiles)
  int rowTile = tile >> 3;
  int colTile = tile & 7;
  if (rowTile >= nRowTiles) return;        // wave-uniform

  int row0 = rowTile * 16;
  int col0 = colTile * 16;
  int lm = lane & 15;
  int khalf = (lane >= 16) ? 2 : 0;

  const float* aBase = mean + (size_t)(row0 + lm) * DIM + khalf;
  const float* xBase = xdst + (size_t)(row0 + lm) * DIM + khalf;
  const float* wlBase = Wl + (size_t)(col0 + lm) * DIM + khalf;
  const float* wrBase = Wr + (size_t)(col0 + lm) * DIM + khalf;

  float bval = bias[col0 + lm];
  v8f accA = {bval, bval, bval, bval, bval, bval, bval, bval};
  v8f accB = {0.f, 0.f, 0.f, 0.f, 0.f, 0.f, 0.f, 0.f};

#pragma unroll 4
  for (int kk = 0; kk < DIM; kk += 4) {
    v2f am = *(const v2f*)(aBase + kk);
    v2f bl = *(const v2f*)(wlBase + kk);
    v2f ax = *(const v2f*)(xBase + kk);
    v2f br = *(const v2f*)(wrBase + kk);
    accA = __builtin_amdgcn_wmma_f32_16x16x4_f32(false, am, false, bl, (short)0,
                                                 accA, false, false);
    accB = __builtin_amdgcn_wmma_f32_16x16x4_f32(false, ax, false, br, (short)0,
                                                 accB, false, false);
  }

  float* orow = out + (size_t)(row0 + ((lane >= 16) ? 8 : 0)) * DIM + col0 + lm;
#pragma unroll
  for (int r = 0; r < 8; ++r) {
    float v = accA[r] + accB[r];
    if (relu) v = fmaxf(v, 0.0f);
    orow[(size_t)r * DIM] = v;
  }
}

// -------------------- cosine-similarity decode: one wave per pair --------------------
__global__ __launch_bounds__(256) void k_decode(
    const float* __restrict__ zu, const float* __restrict__ zj,
    const int* __restrict__ elu, const int* __restrict__ elj,
    float* __restrict__ out, int n) {
  int p = blockIdx.x * 8 + (threadIdx.x >> 5);
  if (p >= n) return;
  int lane = threadIdx.x & 31;
  const float4 a = *(const float4*)(zu + (size_t)elu[p] * DIM + lane * 4);
  const float4 b = *(const float4*)(zj + (size_t)elj[p] * DIM + lane * 4);
  float dot = a.x * b.x + a.y * b.y + a.z * b.z + a.w * b.w;
  float na = a.x * a.x + a.y * a.y + a.z * a.z + a.w * a.w;
  float nb = b.x * b.x + b.y * b.y + b.z * b.z + b.w * b.w;
#pragma unroll
  for (int off = 16; off > 0; off >>= 1) {
    dot += __shfl_xor(dot, off, 32);
    na += __shfl_xor(na, off, 32);
    nb += __shfl_xor(nb, off, 32);
  }
  if (lane == 0) {
    out[p] = dot / (fmaxf(sqrtf(na), 1e-8f) * fmaxf(sqrtf(nb), 1e-8f));
  }
}

// -------------------- host launcher --------------------

extern "C" void kernel_launch(void* const* d_in, const int* in_sizes, int n_in,
                              void* d_out, int out_size, void* d_ws, size_t ws_size,
                              hipStream_t stream) {
  const float* x_user = (const float*)d_in[0];
  const float* x_job = (const float*)d_in[1];
  const int* uj_src = (const int*)d_in[2];
  const int* uj_dst = (const int*)d_in[3];
  const int* ju_src = (const int*)d_in[4];
  const int* ju_dst = (const int*)d_in[5];
  const int* el_user = (const int*)d_in[6];
  const int* el_job = (const int*)d_in[7];
  const float* W1l_uj = (const float*)d_in[8];
  const float* W1r_uj = (const float*)d_in[9];
  const float* W1l_ju = (const float*)d_in[10];
  const float* W1r_ju = (const float*)d_in[11];
  const float* W2l_uj = (const float*)d_in[12];
  const float* W2r_uj = (const float*)d_in[13];
  const float* W2l_ju = (const float*)d_in[14];
  const float* W2r_ju = (const float*)d_in[15];
  const float* b1_uj = (const float*)d_in[16];
  const float* b1_ju = (const float*)d_in[17];
  const float* b2_uj = (const float*)d_in[18];
  const float* b2_ju = (const float*)d_in[19];
  float* out = (float*)d_out;

  float* ws = (float*)d_ws;
  float* agg_user = ws; ws += (size_t)NUU * DIM;   // neighbor sums/means for users
  float* agg_job  = ws; ws += (size_t)NJJ * DIM;   // neighbor sums/means for jobs
  float* cnt_user = ws; ws += NUU;
  float* cnt_job  = ws; ws += NJJ;
  float* h_user   = ws; ws += (size_t)NUU * DIM;
  float* h_job    = ws; ws += (size_t)NJJ * DIM;
  float* z_user   = ws; ws += (size_t)NUU * DIM;
  float* z_job    = ws; ws += (size_t)NJJ * DIM;

  const int nUF = NUU * DIM, nJF = NJJ * DIM;
  dim3 blk(256);

  // ---- layer 1 aggregation ----
  k_zero<<<(nJF + 255) / 256, blk, 0, stream>>>(agg_job, nJF);
  k_zero<<<(nUF + 255) / 256, blk, 0, stream>>>(agg_user, nUF);
  k_zero<<<(NJJ + 255) / 256, blk, 0, stream>>>(cnt_job, NJJ);
  k_zero<<<(NUU + 255) / 256, blk, 0, stream>>>(cnt_user, NUU);
  k_count<<<(NEDGE + 255) / 256, blk, 0, stream>>>(uj_dst, cnt_job, NEDGE);
  k_count<<<(NEDGE + 255) / 256, blk, 0, stream>>>(ju_dst, cnt_user, NEDGE);
  k_scatter<<<NEDGE / 8, blk, 0, stream>>>(x_user, uj_src, uj_dst, agg_job, NEDGE);
  k_scatter<<<NEDGE / 8, blk, 0, stream>>>(x_job, ju_src, ju_dst, agg_user, NEDGE);
  k_mean<<<(nJF + 255) / 256, blk, 0, stream>>>(agg_job, cnt_job, nJF);
  k_mean<<<(nUF + 255) / 256, blk, 0, stream>>>(agg_user, cnt_user, nUF);

  // ---- layer 1 GEMMs (WMMA f32, + bias + relu) ----
  k_sage_gemm<<<NJJ / 16, blk, 0, stream>>>(agg_job, x_job, W1l_uj, W1r_uj,
                                            b1_uj, h_job, NJJ / 16, 1);
  k_sage_gemm<<<NUU / 16, blk, 0, stream>>>(agg_user, x_user, W1l_ju, W1r_ju,
                                            b1_ju, h_user, NUU / 16, 1);

  // ---- layer 2 aggregation (counts reused) ----
  k_zero<<<(nJF + 255) / 256, blk, 0, stream>>>(agg_job, nJF);
  k_zero<<<(nUF + 255) / 256, blk, 0, stream>>>(agg_user, nUF);
  k_scatter<<<NEDGE / 8, blk, 0, stream>>>(h_user, uj_src, uj_dst, agg_job, NEDGE);
  k_scatter<<<NEDGE / 8, blk, 0, stream>>>(h_job, ju_src, ju_dst, agg_user, NEDGE);
  k_mean<<<(nJF + 255) / 256, blk, 0, stream>>>(agg_job, cnt_job, nJF);
  k_mean<<<(nUF + 255) / 256, blk, 0, stream>>>(agg_user, cnt_user, nUF);

  // ---- layer 2 GEMMs (no relu) ----
  k_sage_gemm<<<NJJ / 16, blk, 0, stream>>>(agg_job, h_job, W2l_uj, W2r_uj,
                                            b2_uj, z_job, NJJ / 16, 0);
  k_sage_gemm<<<NUU / 16, blk, 0, stream>>>(agg_user, h_user, W2l_ju, W2r_ju,
                                            b2_ju, z_user, NUU / 16, 0);

  // ---- cosine decode ----
  k_decode<<<NEL / 8, blk, 0, stream>>>(z_user, z_job, el_user, el_job, out, NEL);
}